// LGA2_89395449299014
// MI455X (gfx1250) — compile-verified
//
#include <hip/hip_runtime.h>

// ---------------------------------------------------------------------------
// LGA^2 (local guided aggregation applied twice), fused through LDS.
// MI455X / gfx1250. Data movement via CDNA5 Tensor Data Mover (TDM):
//   - one 3-D tensor_load_to_lds for the 25x12x36 filter tile (reused by all
//     64 channels -> 64x fewer filter reads than naive),
//   - double-buffered 2-D tensor_load_to_lds per-channel x tiles, waited with
//     s_wait_tensorcnt and overlapped with the FMA passes.
// Compute is fp32 VALU FMA (WMMA f32 16x16x4 via a banded-matrix embedding
// would run <25% useful MACs plus operand-shuffle overhead -> slower).
// ---------------------------------------------------------------------------

typedef unsigned int       u32;
typedef unsigned long long u64;
typedef u32   u32x4 __attribute__((ext_vector_type(4)));
typedef int   i32x4 __attribute__((ext_vector_type(4)));
typedef int   i32x8 __attribute__((ext_vector_type(8)));
typedef float f32x4 __attribute__((ext_vector_type(4)));
typedef float f32x2 __attribute__((ext_vector_type(2)));

#define N_   2
#define C_   64
#define H_   256
#define W_   512
#define RAD  2
#define KK   5
#define NF   25           // (2r+1)^2
#define TH   8            // output tile height
#define TW   32           // output tile width
#define TMPH (TH + 2*RAD) // 12  temp halo tile
#define TMPW (TW + 2*RAD) // 36
#define XH   (TH + 4*RAD) // 16  x halo tile
#define XW   (TW + 4*RAD) // 40
#define CPB  16           // channels per outer iteration
#define NIT  (C_ / CPB)   // 4

// LDS layout (float offsets). Total 38192 floats = 152768 B -> 2 WGs / WGP.
#define OFF_F 0
#define SZ_F  (NF * TMPH * TMPW)      // 10800
#define OFF_X (OFF_F + SZ_F)
#define SZ_XB (CPB * XH * XW)         // 10240 (x2 buffers)
#define OFF_T (OFF_X + 2 * SZ_XB)     // 31280
#define SZ_T  (CPB * TMPH * TMPW)     // 6912
#define SMEMF (OFF_T + SZ_T)          // 38192

// ---- TDM descriptor build + issue (clang-23 6-arg builtin form) ------------
// Args 1..4 = D# groups 0..3 (4+8+4+4 dwords, per CDNA5 ISA 8.3-8.6); the
// extra int32x8 corresponds to the encoding's unused VADDR4 slot -> zeros.
// Group0: count=1 | lds_addr | global_addr(57b) | type=2
// Group1: data_size=4B, tensor_dim0/1, tile_dim0/1/2, dim strides (elements)
// Group2: tensor_dim2 (3-D); Group3: zeros.
__device__ __forceinline__ void tdm_load_to_lds(
    u32 lds_byte, const float* gptr,
    u32 tile0, u32 tile1, u32 tile2,
    u32 td0, u32 td1, u32 td2,
    u64 s0, u64 s1)
{
  u64 ga = (u64)(const void*)gptr;
  u32x4 g0;
  g0[0] = 1u;                                            // count=1 (user D#)
  g0[1] = lds_byte;                                      // lds_addr
  g0[2] = (u32)ga;                                       // global_addr[31:0]
  g0[3] = (u32)((ga >> 32) & 0x01FFFFFFu) | (2u << 30);  // addr[56:32]|type=2
  i32x8 g1;
  g1[0] = (int)(2u << 16);                               // data_size = 4B
  g1[1] = (int)((td0 & 0xFFFFu) << 16);                  // tensor_dim0 lo
  g1[2] = (int)((td0 >> 16) | ((td1 & 0xFFFFu) << 16));  // td0 hi | td1 lo
  g1[3] = (int)((td1 >> 16) | (tile0 << 16));            // td1 hi | tile_dim0
  g1[4] = (int)((tile1 & 0xFFFFu) | (tile2 << 16));      // tile_dim1|tile_dim2
  g1[5] = (int)(u32)(s0 & 0xFFFFFFFFull);                // dim0 stride lo32
  g1[6] = (int)((u32)((s0 >> 32) & 0xFFFFull) |
                ((u32)(s1 & 0xFFFFull) << 16));          // s0 hi16 | s1 lo16
  g1[7] = (int)(u32)((s1 >> 16) & 0xFFFFFFFFull);        // s1[47:16]
  i32x4 g2; g2[0] = (int)td2; g2[1] = 0; g2[2] = 0; g2[3] = 0;
  i32x4 g3; g3[0] = 0; g3[1] = 0; g3[2] = 0; g3[3] = 0;
  i32x8 g4; g4[0]=0; g4[1]=0; g4[2]=0; g4[3]=0; g4[4]=0; g4[5]=0; g4[6]=0; g4[7]=0;
  __builtin_amdgcn_tensor_load_to_lds(g0, g1, g2, g3, g4, 0);
}

// ---- generic (border-tile) zero-padded loaders -----------------------------
__device__ __forceinline__ void load_x_generic(float* dst, const float* __restrict__ X,
                                               int n, int cb, int th, int tw, int tid)
{
  for (int idx = tid; idx < SZ_XB; idx += 256) {
    int cc = idx / (XH * XW);
    int rm = idx % (XH * XW);
    int r  = rm / XW, q = rm - r * XW;
    int gh = th - 2 * RAD + r;
    int gw = tw - 2 * RAD + q;
    float v = 0.0f;
    if ((unsigned)gh < (unsigned)H_ && (unsigned)gw < (unsigned)W_)
      v = X[(((size_t)(n * C_ + cb + cc)) * H_ + gh) * W_ + gw];
    dst[idx] = v;
  }
}

__global__ __launch_bounds__(256)
void lga2_fused(const float* __restrict__ X, const float* __restrict__ F,
                float* __restrict__ OUT)
{
  extern __shared__ float smem[];
  float* sF = smem + OFF_F;
  float* sT = smem + OFF_T;

  const int tid = threadIdx.x;
  const int wt  = blockIdx.x;
  const int ht  = blockIdx.y;
  const int n   = blockIdx.z;
  const int th  = ht * TH;
  const int tw  = wt * TW;

  // tile whose full x/filter halo is in-bounds -> pure TDM path
  const bool interior = (ht >= 1) && (ht <= H_ / TH - 2) &&
                        (wt >= 1) && (wt <= W_ / TW - 2);
  const u32 ldsBase = __builtin_amdgcn_groupstaticsize();
  const int wv = __builtin_amdgcn_readfirstlane(tid >> 5);   // wave id 0..7

  // -------- prologue: filters (once) + x channels [0,16) into buffer 0 -----
  if (interior) {
    if (tid < 32) {  // wave 0 issues the single 3-D filter-tile DMA (43 KB)
      const float* gF = F + (((size_t)n * NF) * H_ + (th - RAD)) * W_ + (tw - RAD);
      tdm_load_to_lds(ldsBase + OFF_F * 4u, gF,
                      TMPW, TMPH, NF, W_, H_, NF,
                      (u64)W_, (u64)(H_ * W_));
    }
    for (int k = 0; k < 2; ++k) {        // each wave DMAs 2 channel tiles
      int cl = wv * 2 + k;
      const float* gX = X + (((size_t)(n * C_ + cl)) * H_ + (th - 2 * RAD)) * W_
                          + (tw - 2 * RAD);
      tdm_load_to_lds(ldsBase + (u32)(OFF_X + cl * (XH * XW)) * 4u, gX,
                      XW, XH, 0, W_, H_, 0,
                      (u64)W_, (u64)(H_ * W_));
    }
    __builtin_amdgcn_s_wait_tensorcnt(0);
  } else {
    // zero-padded filter tile: OOB filter == 0 also makes OOB temp pixels 0,
    // which reproduces the reference's zero padding of temp in pass 2.
    for (int idx = tid; idx < SZ_F; idx += 256) {
      int t  = idx / (TMPH * TMPW);
      int rm = idx % (TMPH * TMPW);
      int r  = rm / TMPW, q = rm - r * TMPW;
      int gh = th - RAD + r, gw = tw - RAD + q;
      float v = 0.0f;
      if ((unsigned)gh < (unsigned)H_ && (unsigned)gw < (unsigned)W_)
        v = F[(((size_t)n * NF + t) * H_ + gh) * W_ + gw];
      sF[idx] = v;
    }
    load_x_generic(smem + OFF_X, X, n, 0, th, tw, tid);
  }
  __syncthreads();

  int cur = 0;
  for (int it = 0; it < NIT; ++it) {
    const int   cbase = it * CPB;
    const float* xb   = smem + OFF_X + cur * SZ_XB;

    // -------- prefetch next channel block into the other buffer ------------
    if (it + 1 < NIT) {
      const int cb2 = cbase + CPB;
      if (interior) {
        for (int k = 0; k < 2; ++k) {
          int cl = wv * 2 + k;
          const float* gX = X + (((size_t)(n * C_ + cb2 + cl)) * H_
                              + (th - 2 * RAD)) * W_ + (tw - 2 * RAD);
          tdm_load_to_lds(ldsBase + (u32)(OFF_X + (cur ^ 1) * SZ_XB
                                          + cl * (XH * XW)) * 4u, gX,
                          XW, XH, 0, W_, H_, 0,
                          (u64)W_, (u64)(H_ * W_));
        }
      } else {
        load_x_generic(smem + OFF_X + (cur ^ 1) * SZ_XB, X, n, cb2, th, tw, tid);
      }
    }

    // -------- pass 1: temp on 12x36 halo for 16 channels -------------------
    // unit = 4 channels x 4 pixels: 25 f-dwords + 40 x-dwords, 400 FMAs
    for (int u = tid; u < 4 * (TMPH * (TMPW / 4)); u += 256) {
      const int g  = u / (TMPH * (TMPW / 4));
      const int b  = u % (TMPH * (TMPW / 4));
      const int r  = b / (TMPW / 4);
      const int q0 = (b % (TMPW / 4)) * 4;
      const int c0 = g * 4;
      float acc[4][4];
      #pragma unroll
      for (int cc = 0; cc < 4; ++cc)
        #pragma unroll
        for (int dx = 0; dx < 4; ++dx) acc[cc][dx] = 0.0f;

      #pragma unroll
      for (int i = 0; i < KK; ++i) {
        float xr[4][8];                          // sliding 8-wide window / ch
        #pragma unroll
        for (int cc = 0; cc < 4; ++cc) {
          const f32x4* xp = (const f32x4*)&xb[((c0 + cc) * XH + (r + i)) * XW + q0];
          f32x4 a = xp[0], bv = xp[1];           // ds_load_b128 x2
          xr[cc][0]=a[0]; xr[cc][1]=a[1]; xr[cc][2]=a[2]; xr[cc][3]=a[3];
          xr[cc][4]=bv[0]; xr[cc][5]=bv[1]; xr[cc][6]=bv[2]; xr[cc][7]=bv[3];
        }
        #pragma unroll
        for (int j = 0; j < KK; ++j) {
          const f32x4 fv = *(const f32x4*)&sF[(i * KK + j) * (TMPH * TMPW)
                                             + r * TMPW + q0];
          #pragma unroll
          for (int cc = 0; cc < 4; ++cc)
            #pragma unroll
            for (int dx = 0; dx < 4; ++dx)
              acc[cc][dx] += fv[dx] * xr[cc][j + dx];
        }
      }
      #pragma unroll
      for (int cc = 0; cc < 4; ++cc) {
        f32x4 o; o[0]=acc[cc][0]; o[1]=acc[cc][1]; o[2]=acc[cc][2]; o[3]=acc[cc][3];
        *(f32x4*)&sT[((c0 + cc) * TMPH + r) * TMPW + q0] = o;
      }
    }
    __syncthreads();   // temp tile ready

    // -------- pass 2: 8x32 outputs for 16 channels (1 unit / thread) -------
    {
      const int u  = tid;
      const int g  = u / (TH * (TW / 4));
      const int b  = u % (TH * (TW / 4));
      const int r2 = b / (TW / 4);
      const int q2 = (b % (TW / 4)) * 4;
      const int c0 = g * 4;
      float acc[4][4];
      #pragma unroll
      for (int cc = 0; cc < 4; ++cc)
        #pragma unroll
        for (int dx = 0; dx < 4; ++dx) acc[cc][dx] = 0.0f;

      #pragma unroll
      for (int i = 0; i < KK; ++i) {
        float xr[4][8];
        #pragma unroll
        for (int cc = 0; cc < 4; ++cc) {
          const f32x4* tp = (const f32x4*)&sT[((c0 + cc) * TMPH + (r2 + i)) * TMPW + q2];
          f32x4 a = tp[0], bv = tp[1];
          xr[cc][0]=a[0]; xr[cc][1]=a[1]; xr[cc][2]=a[2]; xr[cc][3]=a[3];
          xr[cc][4]=bv[0]; xr[cc][5]=bv[1]; xr[cc][6]=bv[2]; xr[cc][7]=bv[3];
        }
        #pragma unroll
        for (int j = 0; j < KK; ++j) {
          const float* fp = &sF[(i * KK + j) * (TMPH * TMPW)
                                + (r2 + RAD) * TMPW + (q2 + RAD)];
          f32x2 f01 = *(const f32x2*)fp;         // 8B-aligned ds_load_b64
          f32x2 f23 = *(const f32x2*)(fp + 2);
          float fv0 = f01[0], fv1 = f01[1], fv2 = f23[0], fv3 = f23[1];
          #pragma unroll
          for (int cc = 0; cc < 4; ++cc) {
            acc[cc][0] += fv0 * xr[cc][j + 0];
            acc[cc][1] += fv1 * xr[cc][j + 1];
            acc[cc][2] += fv2 * xr[cc][j + 2];
            acc[cc][3] += fv3 * xr[cc][j + 3];
          }
        }
      }
      const int gh = th + r2, gw = tw + q2;
      #pragma unroll
      for (int cc = 0; cc < 4; ++cc) {
        f32x4 o; o[0]=acc[cc][0]; o[1]=acc[cc][1]; o[2]=acc[cc][2]; o[3]=acc[cc][3];
        *(f32x4*)&OUT[(((size_t)(n * C_ + cbase + c0 + cc)) * H_ + gh) * W_ + gw] = o;
      }
    }

    if (interior && (it + 1 < NIT)) __builtin_amdgcn_s_wait_tensorcnt(0);
    __syncthreads();   // prefetch landed; safe to overwrite sT / swap buffers
    cur ^= 1;
  }
}

extern "C" void kernel_launch(void* const* d_in, const int* in_sizes, int n_in,
                              void* d_out, int out_size, void* d_ws, size_t ws_size,
                              hipStream_t stream)
{
  (void)in_sizes; (void)n_in; (void)out_size; (void)d_ws; (void)ws_size;
  const float* x = (const float*)d_in[0];   // [2,64,256,512] fp32
  const float* f = (const float*)d_in[1];   // [2,25,256,512] fp32
  // d_in[2] = radius scalar (fixed at 2 in the reference; compiled in)
  float* out = (float*)d_out;

  dim3 grid(W_ / TW, H_ / TH, N_);          // (16, 32, 2) = 1024 workgroups
  size_t shmem = (size_t)SMEMF * sizeof(float);   // 152768 B
  lga2_fused<<<grid, 256, shmem, stream>>>(x, f, out);
}